// MultiOneHotDenseEncoder_30855045054713
// MI455X (gfx1250) — compile-verified
//
#include <hip/hip_runtime.h>

// ---------------------------------------------------------------------------
// MultiOneHotDenseEncoder for MI455X (gfx1250, wave32)
//
// out[:, 0:30]    = x[:, 2:32]                       (coalesced copy)
// out[:, 30:94]   = W0[bucket0(x[:,0])]              (one-hot fp32 WMMA)
// out[:, 94:158]  = W1[bucket1(x[:,1])]              (one-hot fp32 WMMA)
//
// Memory-bound kernel (~395 MB total traffic -> ~17 us floor @ 23.3 TB/s).
// Embedding gather expressed as exact one-hot GEMM on V_WMMA_F32_16X16X4_F32
// with ballot-based K-chunk skipping so matrix ops stay under the BW budget.
// W tables staged into LDS with GLOBAL_LOAD_ASYNC_TO_LDS_B32 (ASYNCcnt path).
// ---------------------------------------------------------------------------

#define NCOLS    32
#define OUTCOLS  158      // 30 kept + 64 + 64
#define VV       64       // vocab size per table (plus 1 OOV row)
#define KPAD     68       // 65 rows zero-padded to multiple of 4
#define WAVES    8        // 256 threads / 32

typedef float v2f __attribute__((ext_vector_type(2)));
typedef float v8f __attribute__((ext_vector_type(8)));

// Raw LDS byte offset of a __shared__ object: generic -> AS(3) addrspacecast
// (flat LDS aperture addressing keeps the offset in addr[31:0]), then ptrtoint.
__device__ __forceinline__ unsigned lds_offset_of(const void* p) {
    return (unsigned)(unsigned long long)
           (__attribute__((address_space(3))) const void*)p;
}

__device__ __forceinline__ int bucket_of(int id, const int* __restrict__ ids) {
    // jnp.searchsorted(ids, id) (side='left'), then match check, OOV -> VV
    int lo = 0, hi = VV;
    #pragma unroll
    for (int it = 0; it < 6; ++it) {          // 2^6 = 64 elements
        int mid = (lo + hi) >> 1;
        if (ids[mid] < id) lo = mid + 1; else hi = mid;
    }
    int pos_c = lo < (VV - 1) ? lo : (VV - 1);
    bool match = (lo < VV) && (ids[pos_c] == id);
    return match ? pos_c : VV;
}

__global__ __launch_bounds__(256) void
multi_onehot_encoder(const float* __restrict__ x,
                     const float* __restrict__ W0,
                     const float* __restrict__ W1,
                     const int*   __restrict__ ids0,
                     const int*   __restrict__ ids1,
                     float*       __restrict__ out,
                     int ngroups)
{
    __shared__ float w0s[KPAD * 64];
    __shared__ float w1s[KPAD * 64];
    __shared__ int   i0s[VV];
    __shared__ int   i1s[VV];

    const int tid = threadIdx.x;

    // ---- stage tables into LDS via async global->LDS copies ---------------
    {
        const unsigned l0 = lds_offset_of(w0s);
        const unsigned l1 = lds_offset_of(w1s);
        for (int i = tid; i < (VV + 1) * 64; i += 256) {
            unsigned d0 = l0 + (unsigned)i * 4u;
            unsigned d1 = l1 + (unsigned)i * 4u;
            unsigned long long g0 = (unsigned long long)(W0 + i);
            unsigned long long g1 = (unsigned long long)(W1 + i);
            asm volatile("global_load_async_to_lds_b32 %0, %1, off"
                         :: "v"(d0), "v"(g0) : "memory");
            asm volatile("global_load_async_to_lds_b32 %0, %1, off"
                         :: "v"(d1), "v"(g1) : "memory");
        }
        // zero-pad rows 65..67 (keeps padded-K WMMA products exactly zero)
        for (int i = (VV + 1) * 64 + tid; i < KPAD * 64; i += 256) {
            w0s[i] = 0.0f;
            w1s[i] = 0.0f;
        }
        if (tid < VV) { i0s[tid] = ids0[tid]; i1s[tid] = ids1[tid]; }
        asm volatile("s_wait_asynccnt 0" ::: "memory");
    }
    __syncthreads();

    const int lane  = tid & 31;
    const int wv    = tid >> 5;
    const int m     = lane & 15;      // M row (A) or N col (B/C/D) within tile
    const int hi    = lane >> 4;      // 0 = lanes 0-15, 1 = lanes 16-31
    const int khalf = hi * 2;         // A/B K sub-offset for high half

    for (int g = blockIdx.x * WAVES + wv; g < ngroups; g += gridDim.x * WAVES) {
        const int r0 = g * 16;

        // prefetch next group's x tile (speculative; dropped if OOB)
        __builtin_prefetch(&x[(size_t)(g + gridDim.x * WAVES) * 16 * NCOLS], 0, 1);

        // ---- per-row buckets: lanes 0-15 read col 0, lanes 16-31 read col 1
        float rawv = x[(size_t)(r0 + m) * NCOLS + hi];
        int   id   = (int)rintf(rawv);
        int   b    = hi ? bucket_of(id, i1s) : bucket_of(id, i0s);
        int bA0 = __shfl(b, m, 32);        // bucket0[M] for M = lane&15
        int bA1 = __shfl(b, m + 16, 32);   // bucket1[M]

        // ---- kept columns: out[row, 0:30] = x[row, 2:32] ------------------
        #pragma unroll 4
        for (int rr = 0; rr < 16; ++rr) {
            const int row = r0 + rr;
            float v = x[(size_t)row * NCOLS + lane];
            if (lane >= 2) out[(size_t)row * OUTCOLS + (lane - 2)] = v;
        }

        // ---- embeddings: exact one-hot fp32 WMMA, K-chunk skipping --------
        #pragma unroll
        for (int t = 0; t < 2; ++t) {
            const float* ws = t ? w1s : w0s;
            const int    bb = t ? bA1 : bA0;

            v8f c0 = {}, c1 = {}, c2 = {}, c3 = {};
            for (int kc = 0; kc < KPAD; kc += 4) {
                // wave-uniform skip: does any of the 16 rows hit this K chunk?
                if (__ballot(bb >= kc && bb < kc + 4) == 0ull) continue;

                // A tile (16x4 one-hot): VGPR0 = K kc+khalf, VGPR1 = +1
                v2f a;
                a.x = (bb == kc + khalf)     ? 1.0f : 0.0f;
                a.y = (bb == kc + khalf + 1) ? 1.0f : 0.0f;

                // B tiles (4x16 of W): lane holds N = m, VGPR0/1 = K rows
                const int rka = (kc + khalf) * 64 + m;   // row K, col N
                const int rkb = rka + 64;                // row K+1
                v2f b0, b1, b2, b3;
                b0.x = ws[rka];      b0.y = ws[rkb];
                b1.x = ws[rka + 16]; b1.y = ws[rkb + 16];
                b2.x = ws[rka + 32]; b2.y = ws[rkb + 32];
                b3.x = ws[rka + 48]; b3.y = ws[rkb + 48];

                c0 = __builtin_amdgcn_wmma_f32_16x16x4_f32(false, a, false, b0,
                                                           (short)0, c0, false, false);
                c1 = __builtin_amdgcn_wmma_f32_16x16x4_f32(false, a, false, b1,
                                                           (short)0, c1, false, false);
                c2 = __builtin_amdgcn_wmma_f32_16x16x4_f32(false, a, false, b2,
                                                           (short)0, c2, false, false);
                c3 = __builtin_amdgcn_wmma_f32_16x16x4_f32(false, a, false, b3,
                                                           (short)0, c3, false, false);
            }

            // D layout: lane -> N=m, VGPR v -> row M = v + 8*hi
            const int colbase = 30 + t * VV + m;
            #pragma unroll
            for (int v = 0; v < 8; ++v) {
                const int row = r0 + v + 8 * hi;
                const size_t o = (size_t)row * OUTCOLS + colbase;
                out[o]      = c0[v];
                out[o + 16] = c1[v];
                out[o + 32] = c2[v];
                out[o + 48] = c3[v];
            }
        }
    }
}

extern "C" void kernel_launch(void* const* d_in, const int* in_sizes, int n_in,
                              void* d_out, int out_size, void* d_ws, size_t ws_size,
                              hipStream_t stream) {
    const float* x    = (const float*)d_in[0];
    const float* W0   = (const float*)d_in[1];
    const float* W1   = (const float*)d_in[2];
    const int*   ids0 = (const int*)d_in[3];
    const int*   ids1 = (const int*)d_in[4];
    float*       out  = (float*)d_out;

    const int nrows   = in_sizes[0] / NCOLS;   // 524288
    const int ngroups = nrows / 16;            // 32768

    dim3 grid(1024), block(256);
    hipLaunchKernelGGL(multi_onehot_encoder, grid, block, 0, stream,
                       x, W0, W1, ids0, ids1, out, ngroups);
}